// GIN_58171037057290
// MI455X (gfx1250) — compile-verified
//
#include <hip/hip_runtime.h>

// ---------------------------------------------------------------------------
// GIN (4 stacked GINConv layers, eps=0) for MI455X / gfx1250.
//
// Key change vs round 1: the edge list is layer-invariant, so build CSR by dst
// ONCE (counting sort: count + scan + fill, ~1.6M int atomics total), then
// each layer is a single fused kernel:
//     gather agg tile (16 nodes) into LDS  ->  f32-WMMA MLP  ->  store x_next
// This removes all 4 x 102M f32 atomics of the scatter formulation; the
// per-layer gather is 410MB of coalesced reads served by the 192MB L2
// (x is 25.6MB). GEMMs stay exact-f32 via V_WMMA_F32_16X16X4_F32.
// ---------------------------------------------------------------------------

#define N_NODES 50000
#define N_EDGES 800000
#define DIM     128
#define NLAYERS 4
#define LH_STRIDE 132          // padded LDS row stride (dwords): bank = (4m+k)%64
#define CHUNK   512
#define NCHUNK  ((N_NODES + CHUNK - 1) / CHUNK)   // 98

typedef float v2f __attribute__((ext_vector_type(2)));
typedef float v8f __attribute__((ext_vector_type(8)));

// ---------------- CSR construction --------------------------------------- //

__global__ __launch_bounds__(256) void gin_zero_kernel(int* __restrict__ p, int n)
{
    int i = blockIdx.x * blockDim.x + threadIdx.x;
    if (i < n) p[i] = 0;
}

__global__ __launch_bounds__(256) void gin_count_kernel(
    const long long* __restrict__ edst, int* __restrict__ counts, int n_edges)
{
    int e = blockIdx.x * blockDim.x + threadIdx.x;
    if (e >= n_edges) return;
    __hip_atomic_fetch_add(&counts[(int)edst[e]], 1,
                           __ATOMIC_RELAXED, __HIP_MEMORY_SCOPE_AGENT);
}

// Per-chunk exclusive scan (Hillis-Steele over 512), emit chunk totals.
__global__ __launch_bounds__(CHUNK) void gin_scan1_kernel(
    const int* __restrict__ counts, int* __restrict__ rowStart,
    int* __restrict__ chunkSums)
{
    __shared__ int tmp[CHUNK];
    const int tid = threadIdx.x;
    const int i   = blockIdx.x * CHUNK + tid;
    const int v   = (i < N_NODES) ? counts[i] : 0;
    tmp[tid] = v;
    __syncthreads();
#pragma unroll
    for (int off = 1; off < CHUNK; off <<= 1) {
        int t = (tid >= off) ? tmp[tid - off] : 0;
        __syncthreads();
        tmp[tid] += t;
        __syncthreads();
    }
    if (i < N_NODES) rowStart[i] = tmp[tid] - v;          // exclusive in chunk
    if (tid == CHUNK - 1) chunkSums[blockIdx.x] = tmp[tid];
}

// Exclusive scan of the 98 chunk totals (single block of 128).
__global__ __launch_bounds__(128) void gin_scan2_kernel(
    const int* __restrict__ chunkSums, int* __restrict__ chunkOffs)
{
    __shared__ int tmp[128];
    const int tid = threadIdx.x;
    const int v   = (tid < NCHUNK) ? chunkSums[tid] : 0;
    tmp[tid] = v;
    __syncthreads();
#pragma unroll
    for (int off = 1; off < 128; off <<= 1) {
        int t = (tid >= off) ? tmp[tid - off] : 0;
        __syncthreads();
        tmp[tid] += t;
        __syncthreads();
    }
    chunkOffs[tid] = tmp[tid] - v;
}

// Add chunk offsets; also initialize the fill cursors.
__global__ __launch_bounds__(CHUNK) void gin_scan3_kernel(
    int* __restrict__ rowStart, const int* __restrict__ chunkOffs,
    int* __restrict__ cursor)
{
    int i = blockIdx.x * CHUNK + threadIdx.x;
    if (i < N_NODES) {
        int s = rowStart[i] + chunkOffs[blockIdx.x];
        rowStart[i] = s;
        cursor[i]   = s;
    }
}

__global__ __launch_bounds__(256) void gin_fill_kernel(
    const long long* __restrict__ esrc, const long long* __restrict__ edst,
    int* __restrict__ cursor, int* __restrict__ sortedSrc, int n_edges)
{
    int e = blockIdx.x * blockDim.x + threadIdx.x;
    if (e >= n_edges) return;
    int d   = (int)edst[e];
    int pos = __hip_atomic_fetch_add(&cursor[d], 1,
                                     __ATOMIC_RELAXED, __HIP_MEMORY_SCOPE_AGENT);
    sortedSrc[pos] = (int)esrc[e];
}

// ---------------- fused layer: gather + relu(agg@W1+b1)@W2+b2 ------------- //
// Block = 16 nodes, 8 waves; wave w computes output columns [16w, 16w+16).
// f32 WMMA fragment layout (ISA 7.12.2):
//   A 16x4 : lane L (M=L&15): v0=A[M][k0+2*hi], v1=A[M][k0+2*hi+1], hi=L>>4
//   B 4x16 : lane L (N=L&15): v0=B[k0+2*hi][N], v1=B[k0+2*hi+1][N]
//   C/D    : lane L (N=L&15): vr = D[r + 8*hi][N]
__global__ __launch_bounds__(256) void gin_layer_kernel(
    const float* __restrict__ x,
    const int* __restrict__ rowStart, const int* __restrict__ counts,
    const int* __restrict__ sortedSrc,
    const float* __restrict__ W1, const float* __restrict__ b1,
    const float* __restrict__ W2, const float* __restrict__ b2,
    float* __restrict__ out)
{
    __shared__ float la[16 * LH_STRIDE];
    __shared__ float lh[16 * LH_STRIDE];

    const int wave  = threadIdx.x >> 5;
    const int lane  = threadIdx.x & 31;
    const int n     = lane & 15;
    const int hi    = lane >> 4;
    const int ncol0 = wave * 16;
    const int node0 = blockIdx.x * 16;

    // ---- phase 0: gather agg rows into LDS (one wave per node, 2 nodes/wave)
    for (int m = wave; m < 16; m += 8) {
        const int g  = node0 + m;
        const int s0 = rowStart[g];
        const int cn = counts[g];
        float4 acc = *(const float4*)(x + (size_t)g * DIM + lane * 4);
        for (int j = 0; j < cn; ++j) {
            const int s = sortedSrc[s0 + j];
            const float4 v = *(const float4*)(x + (size_t)s * DIM + lane * 4);
            acc.x += v.x; acc.y += v.y; acc.z += v.z; acc.w += v.w;
        }
        *(float4*)(&la[m * LH_STRIDE + lane * 4]) = acc;   // 16B-aligned: 528|16
    }
    __syncthreads();

    // ---- GEMM1: h = relu(agg @ W1 + b1)
    v8f c = {};
#pragma unroll
    for (int k0 = 0; k0 < DIM; k0 += 4) {
        const int ka = k0 + hi * 2;
        v2f a, b;
        a.x = la[n * LH_STRIDE + ka];
        a.y = la[n * LH_STRIDE + ka + 1];
        b.x = W1[(size_t)ka * DIM + ncol0 + n];
        b.y = W1[(size_t)(ka + 1) * DIM + ncol0 + n];
        c = __builtin_amdgcn_wmma_f32_16x16x4_f32(
                false, a, false, b, (short)0, c, false, false);
    }
    const float bias1 = b1[ncol0 + n];
#pragma unroll
    for (int r = 0; r < 8; ++r) {
        const int m = r + hi * 8;
        float v = c[r] + bias1;
        lh[m * LH_STRIDE + ncol0 + n] = v > 0.0f ? v : 0.0f;
    }
    __syncthreads();

    // ---- GEMM2: out = h @ W2 + b2
    v8f d = {};
#pragma unroll
    for (int k0 = 0; k0 < DIM; k0 += 4) {
        const int ka = k0 + hi * 2;
        v2f a, b;
        a.x = lh[n * LH_STRIDE + ka];
        a.y = lh[n * LH_STRIDE + ka + 1];
        b.x = W2[(size_t)ka * DIM + ncol0 + n];
        b.y = W2[(size_t)(ka + 1) * DIM + ncol0 + n];
        d = __builtin_amdgcn_wmma_f32_16x16x4_f32(
                false, a, false, b, (short)0, d, false, false);
    }
    const float bias2 = b2[ncol0 + n];
#pragma unroll
    for (int r = 0; r < 8; ++r) {
        const int m = r + hi * 8;
        out[(size_t)(node0 + m) * DIM + ncol0 + n] = d[r] + bias2;
    }
}

// ---------------------------------------------------------------------------
extern "C" void kernel_launch(void* const* d_in, const int* in_sizes, int n_in,
                              void* d_out, int out_size, void* d_ws, size_t ws_size,
                              hipStream_t stream)
{
    const float*     x   = (const float*)d_in[0];
    const long long* ei  = (const long long*)d_in[1];   // int64 (2, N_EDGES)
    // d_in[2] = batch (unused: single graph)
    const float*     W1  = (const float*)d_in[3];       // (4,128,128)
    const float*     b1  = (const float*)d_in[4];       // (4,128)
    const float*     W2  = (const float*)d_in[5];
    const float*     b2  = (const float*)d_in[6];
    float*           out = (float*)d_out;

    const long long* esrc = ei;
    const long long* edst = ei + N_EDGES;

    // ---- workspace layout
    float* xbufA     = (float*)d_ws;                         // 6.4M floats
    float* xbufB     = xbufA + (size_t)N_NODES * DIM;        // 6.4M floats
    int*   counts    = (int*)(xbufB + (size_t)N_NODES * DIM);// 50000
    int*   rowStart  = counts   + N_NODES;                   // 50000
    int*   cursor    = rowStart + N_NODES;                   // 50000
    int*   chunkSums = cursor   + N_NODES;                   // 128
    int*   chunkOffs = chunkSums + 128;                      // 128
    int*   sortedSrc = chunkOffs + 128;                      // 800000

    // ---- one-time CSR build (recomputed every call: deterministic work)
    gin_zero_kernel <<<(N_NODES + 255) / 256, 256, 0, stream>>>(counts, N_NODES);
    gin_count_kernel<<<(N_EDGES + 255) / 256, 256, 0, stream>>>(edst, counts, N_EDGES);
    gin_scan1_kernel<<<NCHUNK, CHUNK, 0, stream>>>(counts, rowStart, chunkSums);
    gin_scan2_kernel<<<1, 128, 0, stream>>>(chunkSums, chunkOffs);
    gin_scan3_kernel<<<NCHUNK, CHUNK, 0, stream>>>(rowStart, chunkOffs, cursor);
    gin_fill_kernel <<<(N_EDGES + 255) / 256, 256, 0, stream>>>(
        esrc, edst, cursor, sortedSrc, N_EDGES);

    // ---- 4 fused layers, ping-pong x buffers (gather reads xin, writes xout)
    const int layer_grid = N_NODES / 16;                     // 3125
    const float* xin = x;
    for (int l = 0; l < NLAYERS; ++l) {
        float* xout = (l == NLAYERS - 1) ? out : ((l & 1) ? xbufB : xbufA);
        gin_layer_kernel<<<layer_grid, 256, 0, stream>>>(
            xin, rowStart, counts, sortedSrc,
            W1 + (size_t)l * DIM * DIM, b1 + (size_t)l * DIM,
            W2 + (size_t)l * DIM * DIM, b2 + (size_t)l * DIM,
            xout);
        xin = xout;
    }
}